// LearnableTD_20452634264102
// MI455X (gfx1250) — compile-verified
//
#include <hip/hip_runtime.h>
#include <hip/hip_bf16.h>
#include <math.h>

// Affine map composition for the reverse TD(lambda) recurrence:
//   v_t = a_t * v_{t+1} + b_t,  a_t = g*l_t,  b_t = g*(1 - l_t)
// compose(p, x) applies x AFTER p:  (x.a*p.a, x.a*p.b + x.b)

typedef __attribute__((ext_vector_type(2))) float v2f;
typedef __attribute__((ext_vector_type(8))) float v8f;

#define TD_THREADS 256
#define TD_ITEMS   4
#define TD_CHUNK   (TD_THREADS * TD_ITEMS)   // 1024 elements per block
#define TD_MAXCHUNKS 64                      // 65536 / 1024

__device__ __forceinline__ float td_gamma(const float* __restrict__ rg) {
    return fmaxf(tanhf(rg[0]), 0.0f);
}

__device__ __forceinline__ void td_elem_transform(const float* __restrict__ raw_lambd,
                                                  int t, int N, float g,
                                                  float& xa, float& xb) {
    xa = 1.0f; xb = 0.0f;                    // identity for out-of-range (tail) elems
    if (t < N) {
        float l = fmaxf(tanhf(raw_lambd[t]), 0.0f);
        xa = g * l;
        xb = g * (1.0f - l);
    }
}

// ---------------- Phase 1: per-chunk composed affine transform ----------------
__global__ void td_chunk_transforms(const float* __restrict__ raw_gamma,
                                    const float* __restrict__ raw_lambd,
                                    int N,
                                    float* __restrict__ chunkA,
                                    float* __restrict__ chunkB) {
    __shared__ float sA[TD_THREADS], sB[TD_THREADS];
    const int tid  = threadIdx.x;
    const int base = blockIdx.x * TD_CHUNK;
    const float g  = td_gamma(raw_gamma);

    // Thread-serial compose over TD_ITEMS elements (reverse time order: j asc == t desc)
    float pa = 1.0f, pb = 0.0f;
#pragma unroll
    for (int k = 0; k < TD_ITEMS; ++k) {
        int j = tid * TD_ITEMS + k;
        int t = base + (TD_CHUNK - 1 - j);
        float xa, xb;
        td_elem_transform(raw_lambd, t, N, g, xa, xb);
        float na = xa * pa;
        float nb = fmaf(xa, pb, xb);
        pa = na; pb = nb;
    }

    // Block-wide Hillis-Steele scan of thread-segment transforms (LDS, 8 waves)
    sA[tid] = pa; sB[tid] = pb;
    __syncthreads();
    for (int off = 1; off < TD_THREADS; off <<= 1) {
        float qa = 1.0f, qb = 0.0f;
        if (tid >= off) { qa = sA[tid - off]; qb = sB[tid - off]; }
        __syncthreads();
        if (tid >= off) {
            float xa = sA[tid], xb = sB[tid];
            sA[tid] = xa * qa;
            sB[tid] = fmaf(xa, qb, xb);
        }
        __syncthreads();
    }
    if (tid == TD_THREADS - 1) {
        chunkA[blockIdx.x] = sA[tid];
        chunkB[blockIdx.x] = sB[tid];
    }
}

// ---------- Phase 2: sequential compose across chunks (64 steps, trivial) ----------
__global__ void td_chunk_entries(const float* __restrict__ chunkA,
                                 const float* __restrict__ chunkB,
                                 int nc,
                                 float* __restrict__ ventry) {
    if (threadIdx.x == 0 && blockIdx.x == 0) {
        float v = 1.0f;                       // v at the far right boundary
        for (int c = nc - 1; c >= 0; --c) {
            ventry[c] = v;                    // value entering chunk c from the right
            v = fmaf(chunkA[c], v, chunkB[c]); // value at left edge of chunk c
        }
    }
}

// ------- Phase 3: recompute transforms, apply entry value, emit w, block sums -------
__global__ void td_apply(const float* __restrict__ raw_gamma,
                         const float* __restrict__ raw_lambd,
                         const float* __restrict__ ventry,
                         int N,
                         float* __restrict__ out,
                         float* __restrict__ partials) {
    __shared__ float sA[TD_THREADS], sB[TD_THREADS];
    const int tid  = threadIdx.x;
    const int base = blockIdx.x * TD_CHUNK;
    const float g  = td_gamma(raw_gamma);

    float inca[TD_ITEMS], incb[TD_ITEMS];
    float pa = 1.0f, pb = 0.0f;
#pragma unroll
    for (int k = 0; k < TD_ITEMS; ++k) {
        int j = tid * TD_ITEMS + k;
        int t = base + (TD_CHUNK - 1 - j);
        float xa, xb;
        td_elem_transform(raw_lambd, t, N, g, xa, xb);
        float na = xa * pa;
        float nb = fmaf(xa, pb, xb);
        pa = na; pb = nb;
        inca[k] = pa; incb[k] = pb;
    }

    sA[tid] = pa; sB[tid] = pb;
    __syncthreads();
    for (int off = 1; off < TD_THREADS; off <<= 1) {
        float qa = 1.0f, qb = 0.0f;
        if (tid >= off) { qa = sA[tid - off]; qb = sB[tid - off]; }
        __syncthreads();
        if (tid >= off) {
            float xa = sA[tid], xb = sB[tid];
            sA[tid] = xa * qa;
            sB[tid] = fmaf(xa, qb, xb);
        }
        __syncthreads();
    }

    // Exclusive prefix for this thread's segment
    float ea = 1.0f, eb = 0.0f;
    if (tid > 0) { ea = sA[tid - 1]; eb = sB[tid - 1]; }
    const float ve = ventry[blockIdx.x];

    float lsum = 0.0f;
#pragma unroll
    for (int k = 0; k < TD_ITEMS; ++k) {
        float a = inca[k] * ea;
        float b = fmaf(inca[k], eb, incb[k]);
        int j = tid * TD_ITEMS + k;
        int t = base + (TD_CHUNK - 1 - j);
        if (t < N) {
            float v = fmaf(a, ve, b);
            float w = fmaxf(1.0f - v, 1e-8f);
            out[t] = w;
            lsum += w;
        }
    }

    // Block sum reduction
    __syncthreads();
    sA[tid] = lsum;
    __syncthreads();
    for (int off = TD_THREADS / 2; off > 0; off >>= 1) {
        if (tid < off) sA[tid] += sA[tid + off];
        __syncthreads();
    }
    if (tid == 0) partials[blockIdx.x] = sA[0];
}

// ---- Phase 4a: reduce 64 partials with one V_WMMA_F32_16X16X4_F32, compute scale ----
// A = 16x4 tile holding the 64 partials (2 f32 per lane, bijective), B = ones(4x16)
// => D[i][j] = sum_k A[i][k]  (row sums, same in every column)
// total = sum_i D[i][0] = (lane0: C0..C7) + (lane16: C0..C7)
__global__ void td_scale_compute(const float* __restrict__ partials,
                                 int nc, int N,
                                 float* __restrict__ scale) {
    const int lane = threadIdx.x;            // 32 threads, all active (EXEC all 1s)
    int i0 = 2 * lane, i1 = 2 * lane + 1;
    v2f a;
    a.x = (i0 < nc) ? partials[i0] : 0.0f;
    a.y = (i1 < nc) ? partials[i1] : 0.0f;
    v2f b;
    b.x = 1.0f; b.y = 1.0f;
    v8f c = {};
    c = __builtin_amdgcn_wmma_f32_16x16x4_f32(
        /*neg_a=*/false, a, /*neg_b=*/false, b,
        /*c_mod=*/(short)0, c, /*reuse_a=*/false, /*reuse_b=*/false);
    float s = 0.0f;
#pragma unroll
    for (int r = 0; r < 8; ++r) s += c[r];
    float total = __shfl(s, 0, 32) + __shfl(s, 16, 32);
    if (lane == 0) {
        float mean = total / (float)N;
        scale[0] = 1.0f / fmaxf(mean, 1e-8f);
    }
}

// ---------------- Phase 4b: apply scale ----------------
__global__ void td_scale_apply(float* __restrict__ out,
                               const float* __restrict__ scale,
                               int N) {
    int t = blockIdx.x * blockDim.x + threadIdx.x;
    if (t < N) out[t] *= scale[0];
}

extern "C" void kernel_launch(void* const* d_in, const int* in_sizes, int n_in,
                              void* d_out, int out_size, void* d_ws, size_t ws_size,
                              hipStream_t stream) {
    (void)n_in; (void)out_size; (void)ws_size;
    const float* raw_gamma = (const float*)d_in[0];
    const float* raw_lambd = (const float*)d_in[1];
    float* out = (float*)d_out;
    const int N  = in_sizes[1];                       // 65536
    const int nc = (N + TD_CHUNK - 1) / TD_CHUNK;     // 64 -> exactly one WMMA A-tile

    float* ws       = (float*)d_ws;                   // ~1 KB of scratch
    float* chunkA   = ws;
    float* chunkB   = chunkA + TD_MAXCHUNKS;
    float* ventry   = chunkB + TD_MAXCHUNKS;
    float* partials = ventry + TD_MAXCHUNKS;
    float* scale    = partials + TD_MAXCHUNKS;

    td_chunk_transforms<<<nc, TD_THREADS, 0, stream>>>(raw_gamma, raw_lambd, N,
                                                       chunkA, chunkB);
    td_chunk_entries<<<1, 32, 0, stream>>>(chunkA, chunkB, nc, ventry);
    td_apply<<<nc, TD_THREADS, 0, stream>>>(raw_gamma, raw_lambd, ventry, N,
                                            out, partials);
    td_scale_compute<<<1, 32, 0, stream>>>(partials, nc, N, scale);
    td_scale_apply<<<(N + TD_THREADS - 1) / TD_THREADS, TD_THREADS, 0, stream>>>(out, scale, N);
}